// SSIMLoss_3341484556749
// MI455X (gfx1250) — compile-verified
//
#include <hip/hip_runtime.h>

typedef float v2f __attribute__((ext_vector_type(2)));
typedef float v8f __attribute__((ext_vector_type(8)));

#define WIN 7
#define TILE 64
#define IN_TILE 70           // TILE + WIN - 1
#define XY_STRIDE 72         // 70 + pad
#define H_ROWS 72            // 70 + 2 zero rows so WMMA K-chunks (up to row 71) are safe
#define H_STRIDE 80          // 16*rowdelta keeps half-wave bank sets disjoint
#define IMG 512
#define OUTD 506             // IMG - WIN + 1

__global__ __launch_bounds__(256) void ssim_tile_kernel(
    const float* __restrict__ X, const float* __restrict__ Y,
    const int* __restrict__ drange, float* __restrict__ partials)
{
  extern __shared__ float smem[];
  float* sX = smem;                                   // 70*72
  float* sY = smem + IN_TILE * XY_STRIDE;             // 70*72
  float* hb = smem + 2 * IN_TILE * XY_STRIDE;         // 5 maps * 72*80
  float* hx  = hb;
  float* hy  = hb + 1 * H_ROWS * H_STRIDE;
  float* hxx = hb + 2 * H_ROWS * H_STRIDE;
  float* hyy = hb + 3 * H_ROWS * H_STRIDE;
  float* hxy = hb + 4 * H_ROWS * H_STRIDE;

  const int tid = threadIdx.x;
  const int gr0 = blockIdx.y * TILE;
  const int gc0 = blockIdx.x * TILE;
  const size_t imgOff = (size_t)blockIdx.z * (size_t)(IMG * IMG);
  const float* Xg = X + imgOff;
  const float* Yg = Y + imgOff;

  // Robust data_range decode: works whether harness stored int32(1) or float32(1.0f)
  int iv = drange[0];
  float fv = __int_as_float(iv);
  float d = (fv > 1e-6f && fv < 1e6f) ? fv : (float)iv;
  const float C1 = (0.01f * d) * (0.01f * d);
  const float C2 = (0.03f * d) * (0.03f * d);

  // ---- Stage 1: global -> LDS input tiles (zero-fill OOB) ----
  for (int i = tid; i < IN_TILE * IN_TILE; i += 256) {
    int r = i / IN_TILE, c = i % IN_TILE;
    int gr = gr0 + r, gc = gc0 + c;
    float xv = 0.f, yv = 0.f;
    if (gr < IMG && gc < IMG) {
      xv = Xg[gr * IMG + gc];
      yv = Yg[gr * IMG + gc];
    }
    sX[r * XY_STRIDE + c] = xv;
    sY[r * XY_STRIDE + c] = yv;
  }
  __syncthreads();

  // ---- Stage 2: horizontal 7-tap sums of {x, y, x^2, y^2, x*y} ----
  for (int i = tid; i < H_ROWS * TILE; i += 256) {
    int r = i / TILE, c = i % TILE;
    float sx = 0.f, sy = 0.f, sxx = 0.f, syy = 0.f, sxy = 0.f;
    if (r < IN_TILE) {
      const float* xr = sX + r * XY_STRIDE + c;
      const float* yr = sY + r * XY_STRIDE + c;
#pragma unroll
      for (int k = 0; k < WIN; ++k) {
        float x = xr[k], y = yr[k];
        sx += x; sy += y;
        sxx = fmaf(x, x, sxx);
        syy = fmaf(y, y, syy);
        sxy = fmaf(x, y, sxy);
      }
    }
    int o = r * H_STRIDE + c;
    hx[o] = sx; hy[o] = sy; hxx[o] = sxx; hyy[o] = syy; hxy[o] = sxy;
  }
  __syncthreads();

  // ---- Stage 3: vertical 7-tap sums via V_WMMA_F32_16X16X4_F32 band matmul ----
  // D[m][n] = sum_{r=m}^{m+6} h[rowBase+r][colBase+n], accumulated over 6 K=4 chunks.
  const int lane = tid & 31;
  const int wid  = tid >> 5;
  const int m    = lane & 15;   // A-row (M) and B/D column (N) for this lane
  const int hi   = lane >> 4;   // half-wave selects K pair {0,1} vs {2,3}, D rows j vs j+8
  const float inv49 = 1.f / 49.f;
  const float covn  = 49.f / 48.f;
  float psum = 0.f;

  for (int s = wid; s < 16; s += 8) {           // uniform per wave: 2 sub-tiles each
    int rowBase = (s >> 2) * 16;
    int colBase = (s & 3) * 16;
    v8f a0 = {0,0,0,0,0,0,0,0};
    v8f a1 = {0,0,0,0,0,0,0,0};
    v8f a2 = {0,0,0,0,0,0,0,0};
    v8f a3 = {0,0,0,0,0,0,0,0};
    v8f a4 = {0,0,0,0,0,0,0,0};
#pragma unroll
    for (int chunk = 0; chunk < 6; ++chunk) {
      int r0 = 4 * chunk + (hi ? 2 : 0);        // this lane's two K rows: r0, r0+1
      v2f A;
      A.x = (r0     >= m && r0     <= m + 6) ? 1.f : 0.f;
      A.y = (r0 + 1 >= m && r0 + 1 <= m + 6) ? 1.f : 0.f;
      int bo = (rowBase + r0) * H_STRIDE + colBase + m;
      v2f B;
      B.x = hx[bo];  B.y = hx[bo + H_STRIDE];
      a0 = __builtin_amdgcn_wmma_f32_16x16x4_f32(false, A, false, B, (short)0, a0, false, false);
      B.x = hy[bo];  B.y = hy[bo + H_STRIDE];
      a1 = __builtin_amdgcn_wmma_f32_16x16x4_f32(false, A, false, B, (short)0, a1, false, false);
      B.x = hxx[bo]; B.y = hxx[bo + H_STRIDE];
      a2 = __builtin_amdgcn_wmma_f32_16x16x4_f32(false, A, false, B, (short)0, a2, false, false);
      B.x = hyy[bo]; B.y = hyy[bo + H_STRIDE];
      a3 = __builtin_amdgcn_wmma_f32_16x16x4_f32(false, A, false, B, (short)0, a3, false, false);
      B.x = hxy[bo]; B.y = hxy[bo + H_STRIDE];
      a4 = __builtin_amdgcn_wmma_f32_16x16x4_f32(false, A, false, B, (short)0, a4, false, false);
    }
    // ---- per-element SSIM combine (D layout: VGPR j -> M=j / j+8, N = lane&15) ----
#pragma unroll
    for (int j = 0; j < 8; ++j) {
      int mo   = j + 8 * hi;
      int orow = gr0 + rowBase + mo;
      int ocol = gc0 + colBase + m;
      float ux  = a0[j] * inv49, uy  = a1[j] * inv49;
      float uxx = a2[j] * inv49, uyy = a3[j] * inv49, uxy = a4[j] * inv49;
      float vx  = covn * (uxx - ux * ux);
      float vy  = covn * (uyy - uy * uy);
      float vxy = covn * (uxy - ux * uy);
      float A1 = 2.f * ux * uy + C1;
      float A2 = 2.f * vxy + C2;
      float B1 = ux * ux + uy * uy + C1;
      float B2 = vx + vy + C2;
      float S = (A1 * A2) / (B1 * B2);
      if (orow < OUTD && ocol < OUTD) psum += S;
    }
  }

  // ---- block reduction -> one partial per block ----
  __syncthreads();              // done reading h maps; reuse smem
  smem[tid] = psum;
  __syncthreads();
  for (int st = 128; st > 0; st >>= 1) {
    if (tid < st) smem[tid] += smem[tid + st];
    __syncthreads();
  }
  if (tid == 0) {
    int blin = (blockIdx.z * gridDim.y + blockIdx.y) * gridDim.x + blockIdx.x;
    partials[blin] = smem[0];
  }
}

__global__ __launch_bounds__(256) void ssim_reduce_kernel(
    const float* __restrict__ partials, int n, float invN, float* __restrict__ out)
{
  __shared__ float s[256];
  float acc = 0.f;
  for (int i = threadIdx.x; i < n; i += 256) acc += partials[i];
  s[threadIdx.x] = acc;
  __syncthreads();
  for (int st = 128; st > 0; st >>= 1) {
    if (threadIdx.x < st) s[threadIdx.x] += s[threadIdx.x + st];
    __syncthreads();
  }
  if (threadIdx.x == 0) out[0] = 1.0f - s[0] * invN;
}

extern "C" void kernel_launch(void* const* d_in, const int* in_sizes, int n_in,
                              void* d_out, int out_size, void* d_ws, size_t ws_size,
                              hipStream_t stream) {
  const float* X = (const float*)d_in[0];
  const float* Y = (const float*)d_in[1];
  const int*   dr = (const int*)d_in[2];
  float* out = (float*)d_out;
  float* partials = (float*)d_ws;

  const int B = in_sizes[0] / (IMG * IMG);                  // 64
  const int tilesPerDim = (OUTD + TILE - 1) / TILE;         // 8
  dim3 grid(tilesPerDim, tilesPerDim, B);
  const int nBlocks = tilesPerDim * tilesPerDim * B;        // 4096
  const size_t smemBytes =
      (size_t)(2 * IN_TILE * XY_STRIDE + 5 * H_ROWS * H_STRIDE) * sizeof(float); // 155,520 B

  ssim_tile_kernel<<<grid, 256, smemBytes, stream>>>(X, Y, dr, partials);

  const float invN = 1.0f / ((float)B * (float)OUTD * (float)OUTD);
  ssim_reduce_kernel<<<1, 256, 0, stream>>>(partials, nBlocks, invN, out);
}